// CombinedClDiceDiceFocalLoss_80985903334053
// MI455X (gfx1250) — compile-verified
//
#include <hip/hip_runtime.h>
#include <hip/hip_bf16.h>
#include <stdint.h>

// ---------------- problem constants ----------------
#define BATCH   16
#define HW      262144            // 512*512 = 2^18
#define NPIX    (BATCH * HW)      // 4194304
#define WORDS_PER_IMG 8192        // 512 rows * 16 u32 words (bit-packed)
#define SMOOTH  1e-6f

typedef __attribute__((ext_vector_type(2))) float v2f;
typedef __attribute__((ext_vector_type(8))) float v8f;

#if defined(__has_builtin)
#  if __has_builtin(__builtin_amdgcn_wmma_f32_16x16x4_f32)
#    define HAVE_WMMA_F32X4 1
#  endif
#  if __has_builtin(__builtin_amdgcn_global_load_async_to_lds_b128) && \
      __has_builtin(__builtin_amdgcn_s_wait_asynccnt)
#    define HAVE_ASYNC_LDS 1
#  endif
#  if __has_builtin(__builtin_amdgcn_ballot_w32)
#    define HAVE_BALLOT32 1
#  endif
#endif

// The async-to-LDS builtin takes pointers to 16-byte int vectors:
//   param0: v4i in global (AS1) space, param1: v4i in LDS (AS3) space.
typedef int v4i_gnu __attribute__((vector_size(16)));
typedef __attribute__((address_space(1))) v4i_gnu gv4i_t;
typedef __attribute__((address_space(3))) v4i_gnu lv4i_t;

// ---------------- SWAR helpers ----------------
__device__ __forceinline__ void csa(unsigned &s, unsigned &c,
                                    unsigned a, unsigned b, unsigned d) {
  unsigned t = a ^ b;
  s = t ^ d;
  c = (a & b) | (d & t);
}

__device__ __forceinline__ unsigned ballot32(bool p) {
#if defined(HAVE_BALLOT32)
  return __builtin_amdgcn_ballot_w32(p);
#else
  return (unsigned)__ballot(p);
#endif
}

// One Zhang-Suen sub-iteration on a bit-packed 512x512 image resident in LDS.
// 32 pixels per u32; per-pixel neighbor count B and transition count A are
// computed bit-sliced with carry-save adder trees. Returns nonzero if any
// pixel was removed by this thread.
__device__ int thin_sub(unsigned* img, int tid, int sub) {
  unsigned nw[8];
  int changed = 0;
#pragma unroll
  for (int k = 0; k < 8; ++k) {
    const int q = tid + (k << 10);          // 0..8191, stride 1024
    const int r = q >> 4;                    // row 0..511
    const int w = q & 15;                    // word-in-row 0..15
    const unsigned cu = img[q];
    const unsigned up = (r > 0)               ? img[q - 16] : 0u;
    const unsigned dn = (r < 511)             ? img[q + 16] : 0u;
    const unsigned cl = (w > 0)               ? img[q - 1]  : 0u;
    const unsigned cr = (w < 15)              ? img[q + 1]  : 0u;
    const unsigned ul = (r > 0   && w > 0)    ? img[q - 17] : 0u;
    const unsigned ur = (r > 0   && w < 15)   ? img[q - 15] : 0u;
    const unsigned dl = (r < 511 && w > 0)    ? img[q + 15] : 0u;
    const unsigned dr = (r < 511 && w < 15)   ? img[q + 17] : 0u;
    // Aligned neighbor bit-vectors: bit i = that neighbor of pixel i.
    const unsigned Nn = up, Sn = dn;
    const unsigned En = (cu >> 1) | (cr << 31);
    const unsigned Wn = (cu << 1) | (cl >> 31);
    const unsigned NE = (up >> 1) | (ur << 31);
    const unsigned NW = (up << 1) | (ul >> 31);
    const unsigned SE = (dn >> 1) | (dr << 31);
    const unsigned SW = (dn << 1) | (dl >> 31);
    // ---- B = sum of 8 neighbor bits, bit-sliced (b3 b2 b1 b0) ----
    unsigned s1,c1,s2,c2,s3,c3,s4,c4,u1,u2,v;
    csa(s1,c1, Nn,NE,En); csa(s2,c2, SE,Sn,SW);
    s3 = Wn ^ NW; c3 = Wn & NW;
    csa(s4,c4, s1,s2,s3);
    csa(u1,u2, c1,c2,c3);
    const unsigned b0 = s4;
    const unsigned b1 = u1 ^ c4; v = u1 & c4;
    const unsigned b2 = u2 ^ v;
    const unsigned b3 = u2 & v;
    // 2 <= B <= 6  (B==8 -> b3; B==7 -> b2&b1&b0)
    const unsigned condB = (b1 | b2 | b3) & ~(b3 | (b0 & b1 & b2));
    // ---- A = #(0->1) transitions around N,NE,E,SE,S,SW,W,NW; need A==1 ----
    const unsigned t0=~Nn&NE, t1=~NE&En, t2=~En&SE, t3=~SE&Sn,
                   t4=~Sn&SW, t5=~SW&Wn, t6=~Wn&NW, t7=~NW&Nn;
    csa(s1,c1, t0,t1,t2); csa(s2,c2, t3,t4,t5);
    s3 = t6 ^ t7; c3 = t6 & t7;
    csa(s4,c4, s1,s2,s3);
    csa(u1,u2, c1,c2,c3);
    const unsigned a0 = s4;
    const unsigned a1 = u1 ^ c4; v = u1 & c4;
    const unsigned a2 = u2 ^ v;
    const unsigned a3 = u2 & v;
    const unsigned condA = a0 & ~a1 & ~a2 & ~a3;
    unsigned c3c, c4c;
    if (sub == 0) { c3c = ~(Nn & En & Sn); c4c = ~(En & Sn & Wn); }
    else          { c3c = ~(Nn & En & Wn); c4c = ~(Nn & Sn & Wn); }
    const unsigned rem = cu & condB & condA & c3c & c4c;
    nw[k] = cu & ~rem;
    changed |= (rem != 0u);
  }
  __syncthreads();                          // all reads done before writes
#pragma unroll
  for (int k = 0; k < 8; ++k) img[tid + (k << 10)] = nw[k];
  __syncthreads();
  return changed;
}

// ---------------- kernel 1: pack + in-LDS skeletonize to convergence ----------------
// grid = 32 blocks (batch*2 + which), 1024 threads. Whole image = 32KB LDS.
__global__ void skel_kernel(const float* __restrict__ logits,
                            const int*   __restrict__ tru,
                            unsigned*    __restrict__ skel) {
  __shared__ unsigned img[WORDS_PER_IMG];
  __shared__ int s_changed;
  const int tid   = threadIdx.x;
  const int imgId = blockIdx.x;            // 0..31
  const int b     = imgId >> 1;
  const int which = imgId & 1;             // 0 = pred, 1 = true
  const int lane  = tid & 31;
  const int wv    = tid >> 5;              // 32 waves
  // ---- pack phase: ballot 32 coalesced pixels into one word ----
  for (int s = 0; s < 256; ++s) {
    const int q = wv * 256 + s;
    const int p = q * 32 + lane;
    bool bit;
    if (which == 0) {
      // probs[:,1] > 0.5  <=>  logit1 > logit0
      const size_t base = (size_t)b * (2 * HW) + (size_t)p;
      bit = logits[base + HW] > logits[base];
    } else {
      bit = tru[(size_t)b * HW + (size_t)p] > 0;
    }
    const unsigned word = ballot32(bit);
    if (lane == 0) img[q] = word;
  }
  __syncthreads();
  // ---- thinning to convergence, entirely in LDS ----
  for (int it = 0; it < 512; ++it) {
    if (tid == 0) s_changed = 0;
    __syncthreads();
    int ch = thin_sub(img, tid, 0);
    ch |= thin_sub(img, tid, 1);
    if (ch) s_changed = 1;
    __syncthreads();
    if (!s_changed) break;
  }
  // ---- spill packed skeleton to workspace ----
#pragma unroll
  for (int k = 0; k < 8; ++k) {
    const int q = tid + (k << 10);
    skel[(size_t)imgId * WORDS_PER_IMG + q] = img[q];
  }
}

// ---------------- kernel 2: popcount / clDice per batch ----------------
// Stages packed skeletons into LDS via async global->LDS DMA when available.
__global__ void count_kernel(const unsigned* __restrict__ skel,
                             float* __restrict__ clOut) {
  __shared__ unsigned sp[4096];
  __shared__ unsigned st[4096];
  __shared__ unsigned rp[8], rt[8], ri[8];
  const int b = blockIdx.x, tid = threadIdx.x;
  const int lane = tid & 31, wv = tid >> 5;
  const unsigned* P = skel + (size_t)(2 * b)     * WORDS_PER_IMG;
  const unsigned* T = skel + (size_t)(2 * b + 1) * WORDS_PER_IMG;
  unsigned cp = 0, ct = 0, ci = 0;
  for (int half = 0; half < 2; ++half) {
    const int base = half * 4096;
#if defined(HAVE_ASYNC_LDS)
#pragma unroll
    for (int j = 0; j < 4; ++j) {
      const int c4 = (tid + j * 256) * 4;   // 16B-aligned b128 chunks
      __builtin_amdgcn_global_load_async_to_lds_b128(
          (gv4i_t*)(P + base + c4), (lv4i_t*)(&sp[c4]), 0, 0);
      __builtin_amdgcn_global_load_async_to_lds_b128(
          (gv4i_t*)(T + base + c4), (lv4i_t*)(&st[c4]), 0, 0);
    }
    __builtin_amdgcn_s_wait_asynccnt(0);
    __syncthreads();
#else
    for (int i = tid; i < 4096; i += 256) { sp[i] = P[base + i]; st[i] = T[base + i]; }
    __syncthreads();
#endif
    for (int i = tid; i < 4096; i += 256) {
      const unsigned pw = sp[i], tw = st[i];
      cp += __popc(pw); ct += __popc(tw); ci += __popc(pw & tw);
    }
    __syncthreads();
  }
  for (int o = 16; o >= 1; o >>= 1) {       // wave32 tree
    cp += __shfl_xor(cp, o, 32);
    ct += __shfl_xor(ct, o, 32);
    ci += __shfl_xor(ci, o, 32);
  }
  if (lane == 0) { rp[wv] = cp; rt[wv] = ct; ri[wv] = ci; }
  __syncthreads();
  if (tid == 0) {
    unsigned scp = 0, sct = 0, sci = 0;
    for (int j = 0; j < 8; ++j) { scp += rp[j]; sct += rt[j]; sci += ri[j]; }
    clOut[b] = (2.0f * (float)sci + SMOOTH) / ((float)scp + (float)sct + SMOOTH);
  }
}

// ---------------- exact f32 wave reduction via ones-matrix WMMA ----------------
// D[m][n] = sum_k A[m][k] with B = ones; A per lane = (x, 0) =>
// D[m][*] = x[m] + x[m+16]; per-lane sum of 8 C regs + SWAPX16 add = full sum.
__device__ __forceinline__ float wave_sum(float x) {
#if defined(HAVE_WMMA_F32X4)
  v2f a; a.x = x;    a.y = 0.0f;
  v2f o; o.x = 1.0f; o.y = 1.0f;
  v8f c = {0.f,0.f,0.f,0.f,0.f,0.f,0.f,0.f};
  c = __builtin_amdgcn_wmma_f32_16x16x4_f32(false, a, false, o,
                                            (short)0, c, false, false);
  float s = c[0] + c[1] + c[2] + c[3] + c[4] + c[5] + c[6] + c[7];
  s += __shfl_xor(s, 16, 32);
  return s;
#else
  for (int o = 16; o >= 1; o >>= 1) x += __shfl_xor(x, o, 32);
  return x;
#endif
}

// ---------------- kernel 3: fused focal + dice streaming pass ----------------
// acc[0]=focal_sum acc[1]=inter0 acc[2]=inter1 acc[3]=sum(p1) acc[4]=sum(t)
__global__ void fd_kernel(const float* __restrict__ logits,
                          const int*   __restrict__ tru,
                          float* __restrict__ acc) {
  const int gtid   = blockIdx.x * blockDim.x + threadIdx.x;
  const int stride = gridDim.x * blockDim.x;
  float f = 0.f, i0 = 0.f, i1 = 0.f, sp1 = 0.f, nt1 = 0.f;
  for (int p = gtid; p < NPIX; p += stride) {
    const int b   = p >> 18;                 // HW = 2^18
    const int pix = p & (HW - 1);
    const size_t base = (size_t)b * (2 * HW) + (size_t)pix;
    const float l0 = logits[base];
    const float l1 = logits[base + HW];
    const int   t  = tru[p];
    const float d  = l1 - l0;
    const float p1 = 1.0f / (1.0f + __expf(-d));
    // stable CE: softplus(x) = max(x,0) + log1p(exp(-|x|)), |x| == |d|
    const float ad  = fabsf(d);
    const float tail = __logf(1.0f + __expf(-ad));
    const float x    = (t == 1) ? -d : d;
    const float ce   = fmaxf(x, 0.0f) + tail;
    const float pt   = __expf(-ce);
    const float om   = 1.0f - pt;
    f   += 0.25f * om * om * ce;             // ALPHA=0.25, GAMMA=2
    i0  += (t == 0) ? (1.0f - p1) : 0.0f;
    i1  += (t == 1) ? p1 : 0.0f;
    sp1 += p1;
    nt1 += (float)t;
  }
  // exact f32 wave reduction on the matrix engine (uniform trip count: no divergence)
  f   = wave_sum(f);
  i0  = wave_sum(i0);
  i1  = wave_sum(i1);
  sp1 = wave_sum(sp1);
  nt1 = wave_sum(nt1);
  if ((threadIdx.x & 31) == 0) {
    atomicAdd(&acc[0], f);
    atomicAdd(&acc[1], i0);
    atomicAdd(&acc[2], i1);
    atomicAdd(&acc[3], sp1);
    atomicAdd(&acc[4], nt1);
  }
}

// ---------------- kernel 4: combine ----------------
__global__ void final_kernel(const float* __restrict__ acc,
                             const float* __restrict__ cl,
                             float* __restrict__ out) {
  if (blockIdx.x == 0 && threadIdx.x == 0) {
    float cls = 0.f;
    for (int b = 0; b < BATCH; ++b) cls += cl[b];
    const float cldice_loss = 1.0f - cls / (float)BATCH;
    const float sp = acc[3], nt = acc[4];
    const float card1 = sp + nt;
    const float card0 = 2.0f * (float)NPIX - sp - nt;
    const float dice_loss = 1.0f - 0.5f * ((2.0f * acc[1] + SMOOTH) / (card0 + SMOOTH) +
                                           (2.0f * acc[2] + SMOOTH) / (card1 + SMOOTH));
    const float focal_loss = acc[0] / (float)NPIX;
    out[0] = 0.7f * cldice_loss + 0.1f * dice_loss + 0.2f * focal_loss;
  }
}

__global__ void zero_kernel(float* __restrict__ acc) {
  const int i = threadIdx.x;
  if (i < 5 + BATCH) acc[i] = 0.0f;
}

// ---------------- launcher ----------------
// ws layout: [0, 1MB): 32 bit-packed skeleton images (32 * 8192 u32)
//            then float acc[5] and float cldice[16].
extern "C" void kernel_launch(void* const* d_in, const int* in_sizes, int n_in,
                              void* d_out, int out_size, void* d_ws, size_t ws_size,
                              hipStream_t stream) {
  const float* logits = (const float*)d_in[0];
  const int*   tru    = (const int*)d_in[1];
  float*       out    = (float*)d_out;
  unsigned*    skel   = (unsigned*)d_ws;
  float*       acc    = (float*)((char*)d_ws + (size_t)32 * WORDS_PER_IMG * sizeof(unsigned));
  float*       cl     = acc + 5;

  zero_kernel <<<1,    32,   0, stream>>>(acc);
  skel_kernel <<<32,   1024, 0, stream>>>(logits, tru, skel);
  count_kernel<<<BATCH, 256, 0, stream>>>(skel, cl);
  fd_kernel   <<<2048,  256, 0, stream>>>(logits, tru, acc);
  final_kernel<<<1,     1,   0, stream>>>(acc, cl, out);
  (void)in_sizes; (void)n_in; (void)out_size; (void)ws_size;
}